// Perception3D_17162689314953
// MI455X (gfx1250) — compile-verified
//
#include <hip/hip_runtime.h>

namespace {

constexpr int B = 4, C = 16, D = 64, H = 64, W = 64;
constexpr int G = 5;
constexpr int HWp = H * W;          // 4096 floats = 16 KB per plane
constexpr int HWv = HWp / 4;        // 1024 float4s per plane

// Types matching the async builtin's parameters (global / LDS address spaces).
typedef int v4i __attribute__((vector_size(16)));
typedef v4i __attribute__((address_space(1)))* gas_v4ip;
typedef v4i __attribute__((address_space(3)))* las_v4ip;

// True vector type (HIP float4 is a struct; __builtin_nontemporal_store needs a vector).
typedef float vf4 __attribute__((ext_vector_type(4)));

// Async DMA of 16 bytes global -> LDS, per lane (gfx1250 GLOBAL_LOAD_ASYNC_TO_LDS_B128).
__device__ __forceinline__ void async_b128(const float* g, float* l) {
#if defined(__gfx1250__) && __has_builtin(__builtin_amdgcn_global_load_async_to_lds_b128)
    __builtin_amdgcn_global_load_async_to_lds_b128((gas_v4ip)g, (las_v4ip)l, 0, 0);
#elif defined(__gfx1250__)
    asm volatile("global_load_async_to_lds_b128 %0, %1, off"
                 :: "v"((las_v4ip)l), "v"((gas_v4ip)g)
                 : "memory");
#else
    *(vf4*)l = *(const vf4*)g;
#endif
}

__device__ __forceinline__ void wait_async0() {
#if defined(__gfx1250__) && __has_builtin(__builtin_amdgcn_s_wait_asynccnt)
    __builtin_amdgcn_s_wait_asynccnt(0);
#elif defined(__gfx1250__)
    asm volatile("s_wait_asynccnt 0" ::: "memory");
#endif
}

__global__ __launch_bounds__(256)
void perception3d_kernel(const float* __restrict__ in, float* __restrict__ out) {
    __shared__ float sm[3 * HWp]; // 48 KB: planes d-1, d, d+1

    const int d   = blockIdx.x;
    const int c   = blockIdx.y;
    const int b   = blockIdx.z;
    const int tid = threadIdx.x;

    // Stage the three D-planes into LDS via async-to-LDS DMA; zero-fill OOB planes.
    for (int p = 0; p < 3; ++p) {
        const int dd = d + (p - 1);
        float* lplane = &sm[p * HWp];
        if (dd < 0 || dd >= D) {
            for (int i = tid; i < HWv; i += 256)
                reinterpret_cast<vf4*>(lplane)[i] = (vf4)0.f;
        } else {
            const float* gplane = in + (((size_t)(b * C + c) * D) + dd) * HWp;
#pragma unroll
            for (int i = 0; i < 4; ++i) {
                const int e = (i * 256 + tid) * 4; // float index, 16B aligned
                async_b128(gplane + e, lplane + e);
            }
        }
    }
    wait_async0();
    __syncthreads();

    const vf4* SM4 = reinterpret_cast<const vf4*>(sm);

    // out[b][c*5+g][d][h][w]
    const size_t outBase     = (((size_t)(b * C + c) * G) * D + d) * (size_t)HWp;
    constexpr size_t GSTRIDE = (size_t)D * HWp; // 262144 floats per output channel

#pragma unroll
    for (int it = 0; it < 4; ++it) {
        const int v   = it * 256 + tid;  // float4 index in plane, 0..1023
        const int idx = v * 4;           // float index
        const int h   = v >> 4;          // row (16 float4s per 64-wide row)
        const int w4  = v & 15;          // float4 column

        const vf4 ce = SM4[HWv + v];         // center, plane d
        const vf4 xm = SM4[v];               // plane d-1
        const vf4 xp = SM4[2 * HWv + v];     // plane d+1
        // Speculative reads stay inside sm[0..3*HWp); selects give zero-padding.
        const vf4 yp = (h < H - 1) ? SM4[HWv + v + 16] : (vf4)0.f;
        const vf4 ym = (h > 0)     ? SM4[HWv + v - 16] : (vf4)0.f;
        // W-shifts: register swizzle of center + one scalar DS read at each seam.
        const float zpN = (w4 < 15) ? sm[HWp + idx + 4] : 0.f;
        const float zmP = (w4 > 0)  ? sm[HWp + idx - 1] : 0.f;
        const vf4 zp = {ce.y, ce.z, ce.w, zpN};
        const vf4 zm = {zmP, ce.x, ce.y, ce.z};

        float* o = out + outBase + idx; // 16B aligned
        __builtin_nontemporal_store(ce,                          (vf4*)o);
        __builtin_nontemporal_store(xp + xm + yp + ym + zp + zm, (vf4*)(o + GSTRIDE));
        __builtin_nontemporal_store(xp - xm,                     (vf4*)(o + 2 * GSTRIDE));
        __builtin_nontemporal_store(yp - ym,                     (vf4*)(o + 3 * GSTRIDE));
        __builtin_nontemporal_store(zp - zm,                     (vf4*)(o + 4 * GSTRIDE));
    }
}

} // namespace

extern "C" void kernel_launch(void* const* d_in, const int* in_sizes, int n_in,
                              void* d_out, int out_size, void* d_ws, size_t ws_size,
                              hipStream_t stream) {
    (void)in_sizes; (void)n_in; (void)out_size; (void)d_ws; (void)ws_size;
    const float* in = (const float*)d_in[0];
    float* out = (float*)d_out;
    dim3 grid(D, C, B); // one workgroup per (b, c, d) plane
    perception3d_kernel<<<grid, 256, 0, stream>>>(in, out);
}